// MoE_41609643163845
// MI455X (gfx1250) — compile-verified
//
#include <hip/hip_runtime.h>

// ---------------- problem constants (match reference) ----------------
#define DIMD 512      // model dim
#define MOED 256      // expert hidden dim
#define NEXP 64       // experts
#define NGRP 8        // routing groups
#define TOPG 4        // top groups
#define TOPK 8        // experts per token
#define CAPD 1024     // per-expert capacity
#define TTOK 2048     // tokens (4*512)
static constexpr float ROUTED_SCALE = 2.5f;

typedef float v2f __attribute__((ext_vector_type(2)));
typedef float v8f __attribute__((ext_vector_type(8)));

__device__ __forceinline__ v8f wmma_f32_16x16x4(v2f a, v2f b, v8f c) {
  // 8 args: (neg_a, A, neg_b, B, c_mod, C, reuse_a, reuse_b)
  return __builtin_amdgcn_wmma_f32_16x16x4_f32(false, a, false, b, (short)0, c,
                                               false, false);
}

// ---------------- 1) gating: sigmoid scores + grouped top-k ----------------
__global__ void gate_kernel(const float* __restrict__ x,
                            const float* __restrict__ gw,
                            const float* __restrict__ gb,
                            int* __restrict__ tidx, float* __restrict__ tw) {
  __shared__ float xl[DIMD];
  __shared__ float sc[NEXP];
  const int t = blockIdx.x;
  const int tid = threadIdx.x;  // 64 threads
  const float* xrow = x + (size_t)t * DIMD;
  for (int i = tid; i < DIMD; i += 64) xl[i] = xrow[i];
  __syncthreads();
  float acc = 0.f;
  const float* wrow = gw + (size_t)tid * DIMD;
#pragma unroll 8
  for (int k = 0; k < DIMD; ++k) acc += xl[k] * wrow[k];
  sc[tid] = 1.0f / (1.0f + expf(-acc));
  __syncthreads();
  if (tid == 0) {
    float sb[NEXP];
#pragma unroll
    for (int e = 0; e < NEXP; ++e) sb[e] = sc[e] + gb[e];
    // group score = sum of top-2 within each group of 8
    float gs[NGRP];
    for (int g = 0; g < NGRP; ++g) {
      float m1 = -1e30f, m2 = -1e30f;
      for (int j = 0; j < NEXP / NGRP; ++j) {
        float v = sb[g * (NEXP / NGRP) + j];
        if (v > m1) { m2 = m1; m1 = v; }
        else if (v > m2) m2 = v;
      }
      gs[g] = m1 + m2;
    }
    bool gmask[NGRP];
    for (int g = 0; g < NGRP; ++g) gmask[g] = false;
    for (int s = 0; s < TOPG; ++s) {  // top-4 groups, first-index wins ties
      int bg = 0; float bv = -1e30f;
      for (int g = 0; g < NGRP; ++g)
        if (!gmask[g] && gs[g] > bv) { bv = gs[g]; bg = g; }
      gmask[bg] = true;
    }
    bool taken[NEXP];
    for (int e = 0; e < NEXP; ++e) taken[e] = false;
    int sel[TOPK]; float wv[TOPK]; float wsum = 0.f;
    for (int k = 0; k < TOPK; ++k) {  // top-8 experts in allowed groups
      int be = 0; float bv = -1e30f;
      for (int e = 0; e < NEXP; ++e) {
        if (!gmask[e >> 3] || taken[e]) continue;
        if (sb[e] > bv) { bv = sb[e]; be = e; }
      }
      taken[be] = true;
      sel[k] = be;
      wv[k] = sc[be];  // pre-bias score
      wsum += wv[k];
    }
    const float inv = ROUTED_SCALE / wsum;
    for (int k = 0; k < TOPK; ++k) {
      tidx[(size_t)t * TOPK + k] = sel[k];
      tw[(size_t)t * TOPK + k] = wv[k] * inv;
    }
  }
}

// ---------------- 2) zero expert counters (ws is poisoned) ----------------
__global__ void init_cnt_kernel(int* __restrict__ cnt) {
  if (threadIdx.x < NEXP) cnt[threadIdx.x] = 0;
}

// ---------------- 3) dispatch (build per-expert row lists) ----------------
__global__ void dispatch_kernel(const int* __restrict__ tidx,
                                const float* __restrict__ tw,
                                int* __restrict__ cnt, int* __restrict__ edst,
                                float* __restrict__ ew,
                                float* __restrict__ ybuf) {
  const int i = blockIdx.x * blockDim.x + threadIdx.x;  // (token*K + slot)
  if (i >= TTOK * TOPK) return;
  const int e = tidx[i];
  const int p = atomicAdd(&cnt[e], 1);
  if (p < CAPD) {
    edst[e * CAPD + p] = i;
    ew[e * CAPD + p] = tw[i];
  } else {
    // capacity overflow (~impossible): reference drops -> zero this slot
    float* yr = ybuf + (size_t)i * DIMD;
    for (int d = 0; d < DIMD; ++d) yr[d] = 0.f;
  }
}

// ---------------- 4) grouped SwiGLU FFN via fp32 WMMA, M-tile = 32 --------
// Each workgroup: 32 token-rows of one expert. 8 wave32s. Two 16-row A
// fragments share every B fragment -> 1 global dword per WMMA, two
// independent accumulator chains per B (hides WMMA RAW latency).
__launch_bounds__(256)
__global__ void ffn_kernel(const float* __restrict__ x,
                           const float* __restrict__ w1,
                           const float* __restrict__ w3,
                           const float* __restrict__ w2,
                           const int* __restrict__ cnt,
                           const int* __restrict__ edst,
                           const float* __restrict__ ew,
                           float* __restrict__ ybuf) {
  constexpr int XP = DIMD + 4;  // pad rows: conflict-free column reads
  constexpr int HP = MOED + 4;
  extern __shared__ float smem[];
  float* xs = smem;             // [32][XP]  x tile (32 rows x 512)
  float* hs = smem + 32 * XP;   // [32][HP]  H tile (32 rows x 256)
  __shared__ int rowdst[32];
  __shared__ float roww[32];

  const int e = blockIdx.y;
  const int m0 = blockIdx.x << 5;
  int ne = cnt[e];
  if (ne > CAPD) ne = CAPD;
  if (m0 >= ne) return;  // uniform across block

  const int tid = threadIdx.x;
  if (tid < 32) {
    const int m = m0 + tid;
    if (m < ne) { rowdst[tid] = edst[e * CAPD + m]; roww[tid] = ew[e * CAPD + m]; }
    else        { rowdst[tid] = -1;                 roww[tid] = 0.f; }
  }
  __syncthreads();

  // stage x tile: 4096 float4 over 256 threads, coalesced
#pragma unroll
  for (int j = 0; j < 16; ++j) {
    const int idx = tid + j * 256;    // 0..4095
    const int rr = idx >> 7;          // row 0..31
    const int cc = (idx & 127) << 2;  // col 0..508
    const int dd = rowdst[rr];
    float4 v = make_float4(0.f, 0.f, 0.f, 0.f);
    if (dd >= 0)
      v = *reinterpret_cast<const float4*>(x + (size_t)(dd >> 3) * DIMD + cc);
    *reinterpret_cast<float4*>(&xs[rr * XP + cc]) = v;
  }
  __syncthreads();

  const int wv_id = tid >> 5;   // wave 0..7
  const int lane  = tid & 31;
  const int half  = lane >> 4;  // K half within the 16x16x4 op
  const int l     = lane & 15;  // row (A) / col (B,D) index

  const float* w1e = w1 + (size_t)e * DIMD * MOED;  // [512][256]
  const float* w3e = w3 + (size_t)e * DIMD * MOED;

  // H = silu(X@W1) * (X@W3): 16 N-tiles, 2 per wave, K = 512
#pragma unroll
  for (int sub = 0; sub < 2; ++sub) {
    const int nt = wv_id * 2 + sub;
    const int ncol = nt * 16 + l;
    v8f acc1a = {0.f, 0.f, 0.f, 0.f, 0.f, 0.f, 0.f, 0.f};
    v8f acc1b = acc1a, acc3a = acc1a, acc3b = acc1a;
#pragma unroll 2
    for (int k0 = 0; k0 < DIMD; k0 += 4) {
      const int ka = k0 + half * 2;
      v2f a0; a0.x = xs[l * XP + ka];
              a0.y = xs[l * XP + ka + 1];
      v2f a1; a1.x = xs[(16 + l) * XP + ka];
              a1.y = xs[(16 + l) * XP + ka + 1];
      v2f b1; b1.x = w1e[(size_t)ka * MOED + ncol];
              b1.y = w1e[(size_t)(ka + 1) * MOED + ncol];
      v2f b3; b3.x = w3e[(size_t)ka * MOED + ncol];
              b3.y = w3e[(size_t)(ka + 1) * MOED + ncol];
      acc1a = wmma_f32_16x16x4(a0, b1, acc1a);
      acc1b = wmma_f32_16x16x4(a1, b1, acc1b);
      acc3a = wmma_f32_16x16x4(a0, b3, acc3a);
      acc3b = wmma_f32_16x16x4(a1, b3, acc3b);
    }
    // fused SiLU * gate on accumulator registers; D VGPR r = rows r / r+8
#pragma unroll
    for (int r = 0; r < 8; ++r) {
      const float ga = acc1a[r];
      const float gbv = acc1b[r];
      hs[(r + half * 8) * HP + ncol]      = (ga / (1.f + expf(-ga))) * acc3a[r];
      hs[(16 + r + half * 8) * HP + ncol] = (gbv / (1.f + expf(-gbv))) * acc3b[r];
    }
  }
  __syncthreads();

  // Y = H @ W2: 32 N-tiles, 4 per wave, K = 256
  const float* w2e = w2 + (size_t)e * MOED * DIMD;  // [256][512]
#pragma unroll
  for (int sub = 0; sub < 4; ++sub) {
    const int nt = wv_id * 4 + sub;
    const int ncol = nt * 16 + l;
    v8f acca = {0.f, 0.f, 0.f, 0.f, 0.f, 0.f, 0.f, 0.f};
    v8f accb = acca;
#pragma unroll 2
    for (int k0 = 0; k0 < MOED; k0 += 4) {
      const int ka = k0 + half * 2;
      v2f a0; a0.x = hs[l * HP + ka];
              a0.y = hs[l * HP + ka + 1];
      v2f a1; a1.x = hs[(16 + l) * HP + ka];
              a1.y = hs[(16 + l) * HP + ka + 1];
      v2f b; b.x = w2e[(size_t)ka * DIMD + ncol];
             b.y = w2e[(size_t)(ka + 1) * DIMD + ncol];
      acca = wmma_f32_16x16x4(a0, b, acca);
      accb = wmma_f32_16x16x4(a1, b, accb);
    }
#pragma unroll
    for (int r = 0; r < 8; ++r) {
      const int rowa = r + half * 8;
      const int rowb = 16 + rowa;
      const int da = rowdst[rowa];
      const int db = rowdst[rowb];
      if (da >= 0) ybuf[(size_t)da * DIMD + ncol] = acca[r] * roww[rowa];
      if (db >= 0) ybuf[(size_t)db * DIMD + ncol] = accb[r] * roww[rowb];
    }
  }
}

// ---------------- 5) deterministic combine over the 8 slots ----------------
__global__ void combine_kernel(const float* __restrict__ ybuf,
                               float* __restrict__ out) {
  const int i = blockIdx.x * blockDim.x + threadIdx.x;
  if (i >= TTOK * DIMD) return;
  const int t = i / DIMD;
  const int d = i - t * DIMD;
  const float* base = ybuf + (size_t)t * TOPK * DIMD + d;
  float s = 0.f;
#pragma unroll
  for (int k = 0; k < TOPK; ++k) s += base[(size_t)k * DIMD];
  out[i] = s;
}

extern "C" void kernel_launch(void* const* d_in, const int* in_sizes, int n_in,
                              void* d_out, int out_size, void* d_ws,
                              size_t ws_size, hipStream_t stream) {
  const float* x  = (const float*)d_in[0];
  const float* gw = (const float*)d_in[1];
  const float* gb = (const float*)d_in[2];
  const float* w1 = (const float*)d_in[3];
  const float* w3 = (const float*)d_in[4];
  const float* w2 = (const float*)d_in[5];
  float* out = (float*)d_out;

  char* ws = (char*)d_ws;
  size_t off = 0;
  int*   tidx = (int*)(ws + off);   off += (size_t)TTOK * TOPK * 4;  // 64 KB
  float* tw   = (float*)(ws + off); off += (size_t)TTOK * TOPK * 4;  // 64 KB
  int*   cnt  = (int*)(ws + off);   off += 1024;
  int*   edst = (int*)(ws + off);   off += (size_t)NEXP * CAPD * 4;  // 256 KB
  float* ew   = (float*)(ws + off); off += (size_t)NEXP * CAPD * 4;  // 256 KB
  float* ybuf = (float*)(ws + off);  // T*K*DIM floats = 32 MB

  gate_kernel<<<TTOK, 64, 0, stream>>>(x, gw, gb, tidx, tw);
  init_cnt_kernel<<<1, 64, 0, stream>>>(cnt);
  dispatch_kernel<<<(TTOK * TOPK + 255) / 256, 256, 0, stream>>>(
      tidx, tw, cnt, edst, ew, ybuf);
  constexpr size_t kFfnSmem =
      (size_t)(32 * (DIMD + 4) + 32 * (MOED + 4)) * sizeof(float);  // ~97 KB
  ffn_kernel<<<dim3(CAPD / 32, NEXP), 256, kFfnSmem, stream>>>(
      x, w1, w3, w2, cnt, edst, ew, ybuf);
  combine_kernel<<<(TTOK * DIMD + 255) / 256, 256, 0, stream>>>(ybuf, out);
}